// SubClusteringLoss_4621384810785
// MI455X (gfx1250) — compile-verified
//
#include <hip/hip_runtime.h>
#include <math.h>

typedef float v2f __attribute__((ext_vector_type(2)));
typedef float v8f __attribute__((ext_vector_type(8)));

#define NN 10000
#define KK 16
#define NTILES (NN / 16)      // 625 row tiles
#define NCHUNKS (NN / 4)      // 2500 K-chunks of 4
#define WPB 8                 // waves per block (256 threads, wave32)

// XOR-butterfly add via ds_swizzle_b32 (group-of-32 mode: and=0x1f, or=0, xor=M).
// One DS op per step, no bounds-check cndmask like __shfl_xor generates.
template <int XMASK>
__device__ __forceinline__ float swz_xor_add(float v) {
    int s = __builtin_amdgcn_ds_swizzle(__float_as_int(v), 0x1f | (XMASK << 10));
    return v + __int_as_float(s);
}

// ---------------------------------------------------------------------------
// Kernel 0: zero the three global accumulators [E, sum_sdot, sum_adot]
// ---------------------------------------------------------------------------
__global__ void subclu_init_kernel(float* __restrict__ acc) {
    if (threadIdx.x < 3) acc[threadIdx.x] = 0.0f;
}

// ---------------------------------------------------------------------------
// Kernel 1: per-16-row tile, row dots of adj@out and att@out via f32 WMMA.
//   Each wave takes a cyclic slice of the 2500 K-chunks (unrolled x2) and
//   keeps two f32 16x16 accumulators (adj and att) sharing B loads.
//   sum(adj) is accumulated from the same loaded elements.
// ---------------------------------------------------------------------------
__global__ void __launch_bounds__(256)
subclu_tile_kernel(const float* __restrict__ adj,
                   const float* __restrict__ att,
                   const float* __restrict__ outp,
                   float* __restrict__ out_sdot,   // d_out + 1        [NN]
                   float* __restrict__ out_adot,   // d_out + 1 + NN   [NN]
                   float* __restrict__ acc)        // ws: [E, ssum, asum]
{
    const int tile    = blockIdx.x;
    const int rowbase = tile * 16;
    const int tid     = threadIdx.x;
    const int lane    = tid & 31;
    const int wave    = tid >> 5;     // 0..7
    const int half    = lane >> 4;    // 0 or 1
    const int col     = lane & 15;

    __shared__ float s_sdot[16];
    __shared__ float s_adot[16];
    __shared__ float s_E;
    if (tid < 16) { s_sdot[tid] = 0.0f; s_adot[tid] = 0.0f; }
    if (tid == 0) s_E = 0.0f;
    __syncthreads();

    // A-matrix 16x4 f32 layout (ISA 7.12.2): lanes 0-15 hold K=0,1 of row M=lane;
    // lanes 16-31 hold K=2,3 of row M=lane-16.  -> per-lane b64 load of 2 floats.
    // Default RT hint: each 128B line is reused by 16 requests from the 8 waves
    // of this block at the same iteration index, so keep it in L0.
    const float* arow = adj + (size_t)(rowbase + col) * NN;
    const float* brow = att + (size_t)(rowbase + col) * NN;

    v8f acc_s = {0.f, 0.f, 0.f, 0.f, 0.f, 0.f, 0.f, 0.f};
    v8f acc_a = {0.f, 0.f, 0.f, 0.f, 0.f, 0.f, 0.f, 0.f};
    float esum = 0.0f;

    int chunk = wave;
    // Unrolled x2: 4 WMMAs + 6 loads in flight per iteration.
    for (; chunk + WPB < NCHUNKS; chunk += 2 * WPB) {
        const int kb0 = chunk * 4;
        const int kb1 = kb0 + 4 * WPB;
        const int k00 = kb0 + half * 2;
        const int k01 = kb1 + half * 2;

        v2f as0 = *(const v2f*)(arow + k00);
        v2f aa0 = *(const v2f*)(brow + k00);
        v2f as1 = *(const v2f*)(arow + k01);
        v2f aa1 = *(const v2f*)(brow + k01);

        // B-matrix 4x16 f32 (mirror of A layout).
        v2f b0, b1;
        b0.x = outp[(size_t)(kb0 + 2 * half)     * KK + col];
        b0.y = outp[(size_t)(kb0 + 2 * half + 1) * KK + col];
        b1.x = outp[(size_t)(kb1 + 2 * half)     * KK + col];
        b1.y = outp[(size_t)(kb1 + 2 * half + 1) * KK + col];

        acc_s = __builtin_amdgcn_wmma_f32_16x16x4_f32(
            false, as0, false, b0, (short)0, acc_s, false, false);
        acc_a = __builtin_amdgcn_wmma_f32_16x16x4_f32(
            false, aa0, false, b0, (short)0, acc_a, false, false);
        acc_s = __builtin_amdgcn_wmma_f32_16x16x4_f32(
            false, as1, false, b1, (short)0, acc_s, false, false);
        acc_a = __builtin_amdgcn_wmma_f32_16x16x4_f32(
            false, aa1, false, b1, (short)0, acc_a, false, false);

        esum += (as0.x + as0.y) + (as1.x + as1.y);
    }
    if (chunk < NCHUNKS) {   // odd remainder chunk (waves 0..3)
        const int kb = chunk * 4;
        const int k0 = kb + half * 2;
        v2f a_s = *(const v2f*)(arow + k0);
        v2f a_a = *(const v2f*)(brow + k0);
        v2f b;
        b.x = outp[(size_t)(kb + 2 * half)     * KK + col];
        b.y = outp[(size_t)(kb + 2 * half + 1) * KK + col];
        acc_s = __builtin_amdgcn_wmma_f32_16x16x4_f32(
            false, a_s, false, b, (short)0, acc_s, false, false);
        acc_a = __builtin_amdgcn_wmma_f32_16x16x4_f32(
            false, a_a, false, b, (short)0, acc_a, false, false);
        esum += a_s.x + a_s.y;
    }

    // Reduce esum across the full wave (32 lanes).
    esum = swz_xor_add<16>(swz_xor_add<8>(swz_xor_add<4>(
           swz_xor_add<2>(swz_xor_add<1>(esum)))));
    if (lane == 0) atomicAdd(&s_E, esum);

    // C/D layout: VGPR j, lane l -> row M = j + 8*(l>=16), col N = l&15.
    // Row dot: sum over 16 columns -> XOR butterfly within each 16-lane half.
    #pragma unroll
    for (int j = 0; j < 8; ++j) {
        const int tm = j + 8 * half;                 // tile-local row 0..15
        const float o = outp[(size_t)(rowbase + tm) * KK + col];
        float vs = acc_s[j] * o;
        float va = acc_a[j] * o;
        vs = swz_xor_add<8>(swz_xor_add<4>(swz_xor_add<2>(swz_xor_add<1>(vs))));
        va = swz_xor_add<8>(swz_xor_add<4>(swz_xor_add<2>(swz_xor_add<1>(va))));
        if (col == 0) {
            atomicAdd(&s_sdot[tm], vs);
            atomicAdd(&s_adot[tm], va);
        }
    }
    __syncthreads();

    if (tid == 0) atomicAdd(&acc[0], s_E);
    if (tid < 16) {
        out_sdot[rowbase + tid] = s_sdot[tid];       // raw, normalized later
        out_adot[rowbase + tid] = s_adot[tid];
    }
    if (tid == 32) {
        float ss = 0.0f, sa = 0.0f;
        #pragma unroll
        for (int i = 0; i < 16; ++i) { ss += s_sdot[i]; sa += s_adot[i]; }
        atomicAdd(&acc[1], ss);
        atomicAdd(&acc[2], sa);
    }
}

// ---------------------------------------------------------------------------
// Kernel 2: normalize by E and emit the three [NN] output vectors.
// ---------------------------------------------------------------------------
__global__ void subclu_scale_kernel(float* __restrict__ d_out,
                                    const float* __restrict__ acc) {
    const int i = blockIdx.x * blockDim.x + threadIdx.x;
    if (i >= NN) return;
    const float inv = 1.0f / acc[0];
    const float s = d_out[1 + i] * inv;
    const float a = d_out[1 + NN + i] * inv;
    d_out[1 + i]          = -s;
    d_out[1 + NN + i]     = -a;
    d_out[1 + 2 * NN + i] = -(s + a);
}

// ---------------------------------------------------------------------------
// Kernel 3: cluster-size norm + scalar loss.
//   clu_loss = reg - (sum_sdot + sum_adot)/E,
//   reg = ||colsum(output)|| / N * sqrt(K) - 1
// ---------------------------------------------------------------------------
__global__ void __launch_bounds__(256)
subclu_final_kernel(const float* __restrict__ outp,
                    const float* __restrict__ acc,
                    float* __restrict__ d_out) {
    __shared__ float cs[256];
    const int tid = threadIdx.x;
    const int col = tid & 15;
    const int rg  = tid >> 4;      // 16 row-groups
    float p = 0.0f;
    for (int r = rg; r < NN; r += 16)
        p += outp[(size_t)r * KK + col];
    cs[tid] = p;
    __syncthreads();
    if (tid < 16) {
        float t = 0.0f;
        #pragma unroll
        for (int g = 0; g < 16; ++g) t += cs[g * 16 + tid];
        cs[tid] = t * t;
    }
    __syncthreads();
    if (tid == 0) {
        float nrm2 = 0.0f;
        #pragma unroll
        for (int c = 0; c < 16; ++c) nrm2 += cs[c];
        const float reg = sqrtf(nrm2) / (float)NN * 4.0f - 1.0f;  // sqrt(16)=4
        const float inv = 1.0f / acc[0];
        d_out[0] = reg - (acc[1] + acc[2]) * inv;
    }
}

// ---------------------------------------------------------------------------
extern "C" void kernel_launch(void* const* d_in, const int* in_sizes, int n_in,
                              void* d_out, int out_size, void* d_ws, size_t ws_size,
                              hipStream_t stream) {
    const float* adj  = (const float*)d_in[0];
    const float* att  = (const float*)d_in[1];
    const float* outp = (const float*)d_in[2];
    float* out = (float*)d_out;
    float* acc = (float*)d_ws;   // 3 floats: [E, sum_sdot, sum_adot]

    subclu_init_kernel<<<1, 32, 0, stream>>>(acc);
    subclu_tile_kernel<<<NTILES, 256, 0, stream>>>(
        adj, att, outp, out + 1, out + 1 + NN, acc);
    subclu_scale_kernel<<<(NN + 255) / 256, 256, 0, stream>>>(out, acc);
    subclu_final_kernel<<<1, 256, 0, stream>>>(outp, acc, out);
}